// MSA_Conv2d_1_v1_89773406421553
// MI455X (gfx1250) — compile-verified
//
#include <hip/hip_runtime.h>

typedef float        v2f __attribute__((ext_vector_type(2)));
typedef float        v4f __attribute__((ext_vector_type(4)));
typedef float        v8f __attribute__((ext_vector_type(8)));
typedef unsigned int v4u __attribute__((ext_vector_type(4)));
typedef int          v4i __attribute__((ext_vector_type(4)));
typedef int          v8i __attribute__((ext_vector_type(8)));

#define Bn   16
#define Cn   64
#define Hn   128
#define Wn   128
#define TILE 16
#define KROWS 54      // 3 i-blocks * 18 halo columns (valid K)
#define KPAD  58      // LDS row stride: 54 + 2 zero pad (K=54,55) + 2 (bank pad)
#define KSTEPS 14     // 56 / 4

// One async 4-byte copy global -> LDS (ASYNCcnt, per-lane EXEC).
__device__ __forceinline__ void async_b32(unsigned lds_off, const float* gaddr) {
    asm volatile("global_load_async_to_lds_b32 %0, %1, off"
                 :: "v"(lds_off), "v"(gaddr) : "memory");
}
__device__ __forceinline__ unsigned lds_off(const void* p) {
    return (unsigned)(size_t)p;   // LDS aperture: low 32 bits = LDS byte offset
}

__global__ __launch_bounds__(128)
void msa_conv2d_wmma_f32(const float* __restrict__ Dot,
                         const float* __restrict__ V,
                         float* __restrict__ Out) {
    // V halo tile: 64 channels x (3 rows * 18 cols) flattened, padded to 58
    __shared__ __align__(16) float sV[Cn][KPAD];
    // Banded weight matrix W[56][16] as pairs: sB[p*32 + 2n + q] = W[2p+q][n]
    __shared__ __align__(16) float sB[28 * 32];

    const int x0  = blockIdx.x * TILE;
    const int y   = blockIdx.y;
    const int b   = blockIdx.z;
    const int tid = threadIdx.x;

    // ===== Phase 0: zero-fill (DS stores; disjoint from or equal to DMA dests)
    if (tid < Cn) { v2f z2 = {0.0f, 0.0f}; *(v2f*)&sV[tid][54] = z2; }
    {
        v4f z4 = {0.0f, 0.0f, 0.0f, 0.0f};
        for (int e = tid; e < (28 * 32) / 4; e += 128) ((v4f*)sB)[e] = z4;
    }
    if (y == 0)        for (int e = tid; e < Cn * 18; e += 128) sV[e / 18][e % 18] = 0.0f;
    if (x0 == 0)       for (int e = tid; e < Cn * 3;  e += 128) sV[e / 3][(e % 3) * 18] = 0.0f;

    // ===== Phase 1a: V halo staging
    if (x0 != 0) {
        // TDM: one iterating 2-D tile descriptor moves the whole 13.5 KB halo.
        // Tile (18 cols x tile_dim1 rows) per channel; OOB right/bottom -> zeros.
        if (tid < 32) {                       // wave-level op: issue from wave 0 only
            const int      row0  = (y == 0) ? 0 : y - 1;
            const int      tdim1 = (y == 0) ? 2 : 3;
            const unsigned lds0  = lds_off(&sV[0][0]) + ((y == 0) ? 18u * 4u : 0u);
            const unsigned long long ga =
                (unsigned long long)(size_t)&V[((size_t)b * Cn * Hn + (size_t)row0 * Wn) + (x0 - 1)];

            v4u g0;
            g0[0] = 1u;                                   // count=1, user descriptor
            g0[1] = lds0;                                 // lds_addr (bytes)
            g0[2] = (unsigned)(ga & 0xFFFFFFFFull);       // global_addr[31:0]
            g0[3] = (unsigned)((ga >> 32) & 0x01FFFFFFull) | (2u << 30);  // [56:32] | type=2

            v8i g1;
            g1[0] = (2 << 16) | (1 << 19);     // data_size=4B, iterate_enable=1
            g1[1] = (Wn & 0xFFFF) << 16;       // tensor_dim0[15:0]=128 (bits 63:48)
            g1[2] = (Hn & 0xFFFF) << 16;       // tensor_dim1[15:0]=128 (bits 95:80)
            g1[3] = 18 << 16;                  // tile_dim0=18 (bits 127:112)
            g1[4] = tdim1;                     // tile_dim1 (bits 143:128), tile_dim2=0
            g1[5] = Wn;                        // tensor_dim0_stride[31:0]=128
            g1[6] = (Hn * Wn) << 16;           // tensor_dim1_stride[15:0] (bits 223:208)
            g1[7] = 0;

            v4i g2;                            // iterate mode (channels)
            g2[0] = 0;                         // tensor_dim2 unused (2-D tile)
            g2[1] = KPAD;                      // lds_addr_increment = 58 elements
            g2[2] = Hn * Wn;                   // global_addr_increment = channel stride
            g2[3] = (Cn - 1) << 16;            // iterate_count=63 -> 64 iterations

            v4i g3 = {0, 0, 0, 0};
            v8i g4 = {0, 0, 0, 0, 0, 0, 0, 0}; // extra group (6-arg toolchain), zero
            __builtin_amdgcn_tensor_load_to_lds(g0, g1, g2, g3, g4, 0);
        }
    } else {
        // x0==0: tile starts at col -1 (not expressible in D#) -> per-lane async
        for (int e = tid; e < Cn * KROWS; e += 128) {
            const int ch  = e / KROWS;
            const int rem = e - ch * KROWS;
            const int i   = rem / 18;
            const int m   = rem - i * 18;
            const int row = y + i - 1;
            const int col = x0 + m - 1;
            if (row >= 0 && row < Hn && col >= 0 && col < Wn)
                async_b32(lds_off(&sV[ch][rem]),
                          &V[((b * Cn + ch) * Hn + row) * Wn + col]);
        }
    }

    // ===== Phase 1b: Dot taps scattered into banded matrix: (k=3i+j,n)->W[18i+n+j,n]
    for (int e = tid; e < 9 * TILE; e += 128) {
        const int k  = e >> 4;
        const int n2 = e & 15;
        const int i  = k / 3;
        const int j  = k - 3 * i;
        const int kk = 18 * i + n2 + j;                  // 0..53
        async_b32(lds_off(&sB[(kk >> 1) * 32 + 2 * n2 + (kk & 1)]),
                  &Dot[((b * 9 + k) * Hn + y) * Wn + (x0 + n2)]);
    }

    __builtin_amdgcn_s_wait_tensorcnt(0);
    asm volatile("s_wait_asynccnt 0" ::: "memory");
    __syncthreads();

    // ===== Phase 2: 14 x v_wmma_f32_16x16x4_f32, no per-step VALU
    const int wv   = tid >> 5;
    const int lane = tid & 31;
    const int hi   = lane >> 4;
    const int n    = lane & 15;
    const int c0   = wv * 16;

    const float* aBase = &sV[c0 + n][2 * hi];        // + 4*t per step
    const float* bBase = &sB[hi * 32 + 2 * n];       // + 64*t per step

    v8f acc = {};
#pragma unroll
    for (int t = 0; t < KSTEPS; ++t) {
        v2f a = *(const v2f*)(aBase + 4 * t);
        v2f w = *(const v2f*)(bBase + 64 * t);
        acc = __builtin_amdgcn_wmma_f32_16x16x4_f32(
            false, a, false, w, (short)0, acc, false, false);
    }

    // D: lane/vgpr r -> channel c0 + r + 8*hi, pixel x0 + n (coalesced stores)
#pragma unroll
    for (int r = 0; r < 8; ++r) {
        const int ch = c0 + r + 8 * hi;
        Out[((b * Cn + ch) * Hn + y) * Wn + (x0 + n)] = acc[r];
    }
}

extern "C" void kernel_launch(void* const* d_in, const int* in_sizes, int n_in,
                              void* d_out, int out_size, void* d_ws, size_t ws_size,
                              hipStream_t stream) {
    const float* Dot = (const float*)d_in[0];   // (16, 9, 128, 128) f32
    const float* V   = (const float*)d_in[1];   // (16, 64, 128, 128) f32
    float*       Out = (float*)d_out;           // (16, 64, 128, 128) f32

    dim3 grid(Wn / TILE, Hn, Bn);               // 8 x 128 x 16 tiles
    dim3 block(128);                            // 4 wave32 = 4 channel tiles
    msa_conv2d_wmma_f32<<<grid, block, 0, stream>>>(Dot, V, Out);
}